// Pct_0_37632503448143
// MI455X (gfx1250) — compile-verified
//
#include <hip/hip_runtime.h>
#include <math.h>

// ---------------------------------------------------------------------------
// Problem constants (match the reference)
// ---------------------------------------------------------------------------
#define NPTS   51200
#define C      32
#define KNN    16
#define TILE   2048          // knn LDS tile (2048*3*4 = 24KB)
#define EPSBN  1e-5f
#define INV_SQRT_C 0.17677669529663687f   // 1/sqrt(32)

typedef __attribute__((ext_vector_type(2))) float v2f;
typedef __attribute__((ext_vector_type(8))) float v8f;

// V_WMMA_F32_16X16X4_F32 : D(16x16,f32) = A(16x4,f32) * B(4x16,f32) + C
__device__ __forceinline__ v8f wmma_f32x4(v2f a, v2f b, v8f c) {
  return __builtin_amdgcn_wmma_f32_16x16x4_f32(false, a, false, b, (short)0, c,
                                               false, false);
}

// ---------------------------------------------------------------------------
// Kernel 1: exact KNN (squared euclidean), top-16 incl. self
// 1 thread per query point; candidate coords tiled through LDS.
// ---------------------------------------------------------------------------
__global__ __launch_bounds__(256)
void knn_kernel(const float* __restrict__ coords, int* __restrict__ idx_out) {
  __shared__ float cbuf[TILE * 3];
  const int gid = blockIdx.x * blockDim.x + threadIdx.x;

  const float qx = coords[gid * 3 + 0];
  const float qy = coords[gid * 3 + 1];
  const float qz = coords[gid * 3 + 2];

  float best_d[KNN];
  int   best_i[KNN];
#pragma unroll
  for (int k = 0; k < KNN; ++k) { best_d[k] = 3.4e38f; best_i[k] = 0; }

  for (int base = 0; base < NPTS; base += TILE) {
    // cooperative stage of TILE candidate coords into LDS
    for (int t = threadIdx.x; t < TILE * 3; t += 256)
      cbuf[t] = coords[base * 3 + t];
    // prefetch the next tile while we crunch this one (global_prefetch_b8)
    if (base + TILE < NPTS)
      __builtin_prefetch(coords + (base + TILE) * 3, 0, 1);
    __syncthreads();

    for (int j = 0; j < TILE; ++j) {
      const float dx = qx - cbuf[j * 3 + 0];
      const float dy = qy - cbuf[j * 3 + 1];
      const float dz = qz - cbuf[j * 3 + 2];
      float d2 = dx * dx;
      d2 = fmaf(dy, dy, d2);
      d2 = fmaf(dz, dz, d2);
      if (d2 < best_d[KNN - 1]) {           // rare after warm-up
        float cd = d2; int ci = base + j;   // bubble into sorted list
#pragma unroll
        for (int p = 0; p < KNN; ++p) {
          if (cd < best_d[p]) {
            float td = best_d[p]; int ti = best_i[p];
            best_d[p] = cd; best_i[p] = ci;
            cd = td; ci = ti;
          }
        }
      }
    }
    __syncthreads();
  }

#pragma unroll
  for (int k = 0; k < KNN; ++k) idx_out[gid * KNN + k] = best_i[k];
}

// ---------------------------------------------------------------------------
// Kernel 2: S = (feats @ qw^T + qb) @ kw ; kbq = kb . Q
// 8 waves/block, 16 rows per wave -> 128 rows/block, grid = NPTS/128.
// WMMA f32 16x16x4, K-loop of 8 steps. LDS relayout between the two GEMMs.
// ---------------------------------------------------------------------------
__global__ __launch_bounds__(256)
void qs_kernel(const float* __restrict__ feats,
               const float* __restrict__ qw, const float* __restrict__ qb,
               const float* __restrict__ kw, const float* __restrict__ kb,
               float* __restrict__ S, float* __restrict__ kbq) {
  __shared__ float stage[8][16][C];           // per-wave 16x32 Q tile (16KB)
  const int lane = threadIdx.x & 31;
  const int wave = threadIdx.x >> 5;
  const int m    = lane & 15;                 // row (A) / col (B,C) id
  const int half = lane >> 4;                 // which K-pair / row-half
  const int r0   = blockIdx.x * 128 + wave * 16;

  // ---- GEMM 1: Q = feats @ qw^T  (+ qb) ----
  v8f acc0 = {0,0,0,0,0,0,0,0};
  v8f acc1 = {0,0,0,0,0,0,0,0};
#pragma unroll
  for (int ks = 0; ks < 8; ++ks) {
    const int kk = ks * 4 + half * 2;
    v2f a  = *(const v2f*)(feats + (r0 + m) * C + kk);
    v2f b0 = *(const v2f*)(qw + m * C + kk);          // B[j][c]=qw[c][j]
    v2f b1 = *(const v2f*)(qw + (m + 16) * C + kk);
    acc0 = wmma_f32x4(a, b0, acc0);
    acc1 = wmma_f32x4(a, b1, acc1);
  }
  const float qb0 = qb[m], qb1 = qb[m + 16];
#pragma unroll
  for (int v = 0; v < 8; ++v) { acc0[v] += qb0; acc1[v] += qb1; }

  // ---- C-layout -> LDS (row = v + 8*half, col = m / m+16) ----
#pragma unroll
  for (int v = 0; v < 8; ++v) {
    stage[wave][v + 8 * half][m]      = acc0[v];
    stage[wave][v + 8 * half][m + 16] = acc1[v];
  }
  __syncthreads();

  // ---- kbq[n] = kb . Q[n] ----
  if (lane < 16) {
    float s = 0.f;
#pragma unroll
    for (int c = 0; c < C; ++c) s = fmaf(kb[c], stage[wave][lane][c], s);
    kbq[r0 + lane] = s;
  }

  // ---- GEMM 2: S = Q @ kw ----
  v8f s0 = {0,0,0,0,0,0,0,0};
  v8f s1 = {0,0,0,0,0,0,0,0};
#pragma unroll
  for (int ks = 0; ks < 8; ++ks) {
    const int kk = ks * 4 + half * 2;
    v2f a; a.x = stage[wave][m][kk]; a.y = stage[wave][m][kk + 1];
    v2f b0, b1;                                   // B[c][j]=kw[c][j]
    b0.x = kw[kk * C + m];        b0.y = kw[(kk + 1) * C + m];
    b1.x = kw[kk * C + m + 16];   b1.y = kw[(kk + 1) * C + m + 16];
    s0 = wmma_f32x4(a, b0, s0);
    s1 = wmma_f32x4(a, b1, s1);
  }
#pragma unroll
  for (int v = 0; v < 8; ++v) {
    S[(r0 + v + 8 * half) * C + m]      = s0[v];
    S[(r0 + v + 8 * half) * C + m + 16] = s1[v];
  }
}

// ---------------------------------------------------------------------------
// Kernel 3: per-point attention. One wave per point, lane = channel.
// att[k] = (nbr_k . S[n] + kbq[n]) / sqrt(C); softmax; W[n] = sum_k att*nbr_k
// ---------------------------------------------------------------------------
__global__ __launch_bounds__(256)
void attn_kernel(const float* __restrict__ feats,
                 const float* __restrict__ S, const float* __restrict__ kbq,
                 const int* __restrict__ idx, float* __restrict__ W) {
  const int lane = threadIdx.x & 31;
  const int wave = threadIdx.x >> 5;
  const int n    = blockIdx.x * 8 + wave;

  const float s_val = S[n * C + lane];
  const float kq    = kbq[n];

  float nbrv[KNN];
  float att[KNN];
#pragma unroll
  for (int k = 0; k < KNN; ++k) {
    const int j = idx[n * KNN + k];          // uniform per wave -> scalar load
    nbrv[k] = feats[j * C + lane];
    float p = nbrv[k] * s_val;
#pragma unroll
    for (int off = 16; off >= 1; off >>= 1)  // wave32 all-reduce
      p += __shfl_xor(p, off, 32);
    att[k] = (p + kq) * INV_SQRT_C;
  }

  float mx = att[0];
#pragma unroll
  for (int k = 1; k < KNN; ++k) mx = fmaxf(mx, att[k]);
  float sum = 0.f;
#pragma unroll
  for (int k = 0; k < KNN; ++k) { att[k] = __expf(att[k] - mx); sum += att[k]; }
  const float inv = 1.0f / sum;

  float w = 0.f;
#pragma unroll
  for (int k = 0; k < KNN; ++k) w = fmaf(att[k], nbrv[k], w);
  W[n * C + lane] = w * inv;
}

// ---------------------------------------------------------------------------
// Kernel 4: r = (feats - (W @ vw^T + vb)) @ cw^T + cb  ; BN partial sums
// ---------------------------------------------------------------------------
__global__ __launch_bounds__(256)
void conv_kernel(const float* __restrict__ feats, const float* __restrict__ W,
                 const float* __restrict__ vw, const float* __restrict__ vb,
                 const float* __restrict__ cw, const float* __restrict__ cb,
                 float* __restrict__ r_out, float* __restrict__ stats) {
  __shared__ float stage[8][16][C];
  __shared__ float bsum[C], bsq[C];
  const int lane = threadIdx.x & 31;
  const int wave = threadIdx.x >> 5;
  const int m    = lane & 15;
  const int half = lane >> 4;
  const int r0   = blockIdx.x * 128 + wave * 16;

  if (threadIdx.x < C) { bsum[threadIdx.x] = 0.f; bsq[threadIdx.x] = 0.f; }

  // ---- att_feat = W @ vw^T + vb ----
  v8f acc0 = {0,0,0,0,0,0,0,0};
  v8f acc1 = {0,0,0,0,0,0,0,0};
#pragma unroll
  for (int ks = 0; ks < 8; ++ks) {
    const int kk = ks * 4 + half * 2;
    v2f a  = *(const v2f*)(W + (r0 + m) * C + kk);
    v2f b0 = *(const v2f*)(vw + m * C + kk);
    v2f b1 = *(const v2f*)(vw + (m + 16) * C + kk);
    acc0 = wmma_f32x4(a, b0, acc0);
    acc1 = wmma_f32x4(a, b1, acc1);
  }
  const float vb0 = vb[m], vb1 = vb[m + 16];

  // ---- t = feats - att_feat (in C layout), stage to LDS ----
#pragma unroll
  for (int v = 0; v < 8; ++v) {
    const int row = r0 + v + 8 * half;
    stage[wave][v + 8 * half][m]      = feats[row * C + m]      - (acc0[v] + vb0);
    stage[wave][v + 8 * half][m + 16] = feats[row * C + m + 16] - (acc1[v] + vb1);
  }
  __syncthreads();

  // ---- r = t @ cw^T + cb ----
  v8f c0 = {0,0,0,0,0,0,0,0};
  v8f c1 = {0,0,0,0,0,0,0,0};
#pragma unroll
  for (int ks = 0; ks < 8; ++ks) {
    const int kk = ks * 4 + half * 2;
    v2f a; a.x = stage[wave][m][kk]; a.y = stage[wave][m][kk + 1];
    v2f b0 = *(const v2f*)(cw + m * C + kk);
    v2f b1 = *(const v2f*)(cw + (m + 16) * C + kk);
    c0 = wmma_f32x4(a, b0, c0);
    c1 = wmma_f32x4(a, b1, c1);
  }
  const float cb0 = cb[m], cb1 = cb[m + 16];
  float p0 = 0.f, p1 = 0.f, q0 = 0.f, q1 = 0.f;
#pragma unroll
  for (int v = 0; v < 8; ++v) {
    const float r0v = c0[v] + cb0;
    const float r1v = c1[v] + cb1;
    const int row = r0 + v + 8 * half;
    r_out[row * C + m]      = r0v;
    r_out[row * C + m + 16] = r1v;
    p0 += r0v; q0 = fmaf(r0v, r0v, q0);
    p1 += r1v; q1 = fmaf(r1v, r1v, q1);
  }
  // block-level BN partials via LDS float atomics (ds_add_f32)
  atomicAdd(&bsum[m], p0);      atomicAdd(&bsq[m], q0);
  atomicAdd(&bsum[m + 16], p1); atomicAdd(&bsq[m + 16], q1);
  __syncthreads();
  if (threadIdx.x < C) {
    atomicAdd(&stats[threadIdx.x],     bsum[threadIdx.x]);
    atomicAdd(&stats[C + threadIdx.x], bsq[threadIdx.x]);
  }
}

// ---------------------------------------------------------------------------
// Kernel 5: BN finalize + ReLU + residual
// ---------------------------------------------------------------------------
__global__ __launch_bounds__(256)
void bn_finalize_kernel(const float* __restrict__ feats,
                        const float* __restrict__ r,
                        const float* __restrict__ stats,
                        const float* __restrict__ g, const float* __restrict__ b,
                        float* __restrict__ out) {
  const int gid = blockIdx.x * blockDim.x + threadIdx.x;
  const int c = gid & (C - 1);
  const float mu  = stats[c] * (1.0f / NPTS);
  const float var = stats[C + c] * (1.0f / NPTS) - mu * mu;
  const float rn  = (r[gid] - mu) * __frsqrt_rn(var + EPSBN) * g[c] + b[c];
  out[gid] = feats[gid] + fmaxf(rn, 0.f);
}

__global__ void zero_stats_kernel(float* __restrict__ stats) {
  if (threadIdx.x < 2 * C) stats[threadIdx.x] = 0.f;
}

// ---------------------------------------------------------------------------
// Host launcher
// ---------------------------------------------------------------------------
extern "C" void kernel_launch(void* const* d_in, const int* in_sizes, int n_in,
                              void* d_out, int out_size, void* d_ws, size_t ws_size,
                              hipStream_t stream) {
  const float* feats  = (const float*)d_in[0];
  const float* coords = (const float*)d_in[1];

  // workspace carve-out (256B aligned)
  char* ws = (char*)d_ws;
  size_t off = 0;
  auto carve = [&](size_t bytes) {
    char* p = ws + off;
    off += (bytes + 255) & ~(size_t)255;
    return p;
  };
  int*   idx   = (int*)  carve((size_t)NPTS * KNN * sizeof(int));
  float* Sbuf  = (float*)carve((size_t)NPTS * C * sizeof(float));
  float* kbq   = (float*)carve((size_t)NPTS * sizeof(float));
  float* Wbuf  = (float*)carve((size_t)NPTS * C * sizeof(float));
  float* rbuf  = (float*)carve((size_t)NPTS * C * sizeof(float));
  float* mid   = (float*)carve((size_t)NPTS * C * sizeof(float));
  float* stats = (float*)carve(2 * C * sizeof(float));

  knn_kernel<<<NPTS / 256, 256, 0, stream>>>(coords, idx);

  for (int layer = 0; layer < 2; ++layer) {
    const int base = layer ? 12 : 2;
    const float* qw = (const float*)d_in[base + 0];
    const float* qb = (const float*)d_in[base + 1];
    const float* kw = (const float*)d_in[base + 2];
    const float* kb = (const float*)d_in[base + 3];
    const float* vw = (const float*)d_in[base + 4];
    const float* vb = (const float*)d_in[base + 5];
    const float* cw = (const float*)d_in[base + 6];
    const float* cb = (const float*)d_in[base + 7];
    const float* g  = (const float*)d_in[base + 8];
    const float* bb = (const float*)d_in[base + 9];

    const float* fin  = layer ? (const float*)mid : feats;
    float*       fout = layer ? (float*)d_out : mid;

    zero_stats_kernel<<<1, 64, 0, stream>>>(stats);
    qs_kernel<<<NPTS / 128, 256, 0, stream>>>(fin, qw, qb, kw, kb, Sbuf, kbq);
    attn_kernel<<<NPTS / 8, 256, 0, stream>>>(fin, Sbuf, kbq, idx, Wbuf);
    conv_kernel<<<NPTS / 128, 256, 0, stream>>>(fin, Wbuf, vw, vb, cw, cb,
                                                rbuf, stats);
    bn_finalize_kernel<<<(NPTS * C) / 256, 256, 0, stream>>>(fin, rbuf, stats,
                                                             g, bb, fout);
  }
  (void)in_sizes; (void)n_in; (void)out_size; (void)ws_size;
}